// KVAE_27633819583109
// MI455X (gfx1250) — compile-verified
//
#include <hip/hip_runtime.h>
#include <hip/hip_bf16.h>

// KVAE Kalman smoother for MI455X (gfx1250, wave32).
//
// Kernel 1: mixture expansion  M[bt, 0:96] = softmax(alpha_logits[bt]) @ [A_base|C_base]
//           done as (16-row tiles) x V_WMMA_F32_16X16X4_F32, 6 WMMAs per wave.
// Kernel 2: per-batch forward Kalman filter (information form; only 4x4 inverses,
//           Woodbury identity since R = 0.03*I) fused with backward RTS smoother.
//
// ws layout (floats): mixed[RT*96] | zhat[RT*4] | shatc[RT*16]   (RT = B*T = 524288)
// d_out layout:       z_smooth[RT*4] | std_smooth[RT*16]

#define KB 2048
#define KT 256
#define KDA 20
#define KDZ 4
#define KRT ((size_t)KB * (size_t)KT)

typedef float v2f __attribute__((ext_vector_type(2)));
typedef float v8f __attribute__((ext_vector_type(8)));

// fast hardware reciprocal (v_rcp_f32); adequate for softmax norm / determinant
__device__ __forceinline__ float fast_rcp(float x) {
  return __builtin_amdgcn_rcpf(x);
}

// ---------------------------------------------------------------------------
// Kernel 1: WMMA mixture expansion.
// Rows = (b*T + t) flat index. A-matrix (16x4): row m holds softmax weights
// (alpha0, alpha1, alpha2, 0). B-matrix (4x16 per block): basis columns.
// VGPR layouts per CDNA5 ISA 7.12.2: A frag v[0]={K0|K2}, v[1]={K1|K3};
// B frag mirrored; D frag v[r] = row (r + 8*(lane>>4)), col (lane&15).
// ---------------------------------------------------------------------------
__device__ __forceinline__ float wbasis(const float* __restrict__ Ab,
                                        const float* __restrict__ Cb,
                                        int k, int j) {
  // combined basis row k, column j: j<16 -> A_base[k] flat, else C_base[k] flat
  return (j < 16) ? Ab[k * 16 + j] : Cb[k * 80 + (j - 16)];
}

__global__ void kvae_mix_wmma(const float* __restrict__ alpha_logits, // (RT,3)
                              const float* __restrict__ A_base,       // (3,16)
                              const float* __restrict__ C_base,       // (3,80)
                              float* __restrict__ mixed) {            // (RT,96)
  const int lane = threadIdx.x & 31;
  const size_t wave = (size_t)((blockIdx.x * blockDim.x + threadIdx.x) >> 5);
  const size_t row0 = wave * 16;
  if (row0 >= KRT) return;                 // whole-wave uniform; EXEC stays full

  const int m = lane & 15;                 // tile row owned by this lane
  const int hi = lane >> 4;                // half selector -> K pair
  const size_t row = row0 + (size_t)m;

  // softmax over 3 logits for this lane's row
  const float l0 = alpha_logits[row * 3 + 0];
  const float l1 = alpha_logits[row * 3 + 1];
  const float l2 = alpha_logits[row * 3 + 2];
  const float mx = fmaxf(l0, fmaxf(l1, l2));
  const float e0 = __expf(l0 - mx);
  const float e1 = __expf(l1 - mx);
  const float e2 = __expf(l2 - mx);
  const float inv_s = fast_rcp(e0 + e1 + e2);
  const float a0 = e0 * inv_s, a1 = e1 * inv_s, a2 = e2 * inv_s;

  // A fragment: lanes 0-15 carry (K=0,K=1), lanes 16-31 carry (K=2,K=3=pad0)
  v2f afrag;
  afrag.x = hi ? a2 : a0;
  afrag.y = hi ? 0.0f : a1;

  const int k0 = hi ? 2 : 0;               // K row for frag .x (always valid)
  const int k1ld = hi ? 0 : 1;             // SAFE row to load for frag .y
                                           // (hi half is pad -> value masked to 0,
                                           //  unconditional load avoids EXEC branch)
#pragma unroll
  for (int blk = 0; blk < 6; ++blk) {
    const int n0 = blk * 16;
    const int j = n0 + m;                  // global basis column 0..95
    v2f bfrag;
    bfrag.x = wbasis(A_base, C_base, k0, j);
    const float w1 = wbasis(A_base, C_base, k1ld, j);
    bfrag.y = hi ? 0.0f : w1;              // v_cndmask, no divergent branch
    v8f c = {};
    v8f d = __builtin_amdgcn_wmma_f32_16x16x4_f32(
        /*neg_a=*/false, afrag, /*neg_b=*/false, bfrag,
        /*c_mod=*/(short)0, c, /*reuse_a=*/false, /*reuse_b=*/false);
#pragma unroll
    for (int r = 0; r < 8; ++r) {
      const int mm = r + 8 * hi;           // D row for this VGPR/lane half
      mixed[(row0 + (size_t)mm) * 96 + (size_t)j] = d[r];
    }
  }
}

// ---------------------------------------------------------------------------
// 4x4 dense helpers (row-major, fully unrolled -> registers)
// ---------------------------------------------------------------------------
__device__ __forceinline__ void mul4(const float* A, const float* B, float* O) {
#pragma unroll
  for (int i = 0; i < 4; ++i)
#pragma unroll
    for (int j = 0; j < 4; ++j)
      O[i * 4 + j] = A[i * 4 + 0] * B[0 + j] + A[i * 4 + 1] * B[4 + j] +
                     A[i * 4 + 2] * B[8 + j] + A[i * 4 + 3] * B[12 + j];
}
__device__ __forceinline__ void mul4_ABt(const float* A, const float* B, float* O) {
#pragma unroll
  for (int i = 0; i < 4; ++i)
#pragma unroll
    for (int j = 0; j < 4; ++j)
      O[i * 4 + j] = A[i * 4 + 0] * B[j * 4 + 0] + A[i * 4 + 1] * B[j * 4 + 1] +
                     A[i * 4 + 2] * B[j * 4 + 2] + A[i * 4 + 3] * B[j * 4 + 3];
}
__device__ __forceinline__ void matvec4(const float* A, const float* x, float* y) {
#pragma unroll
  for (int i = 0; i < 4; ++i)
    y[i] = A[i * 4 + 0] * x[0] + A[i * 4 + 1] * x[1] +
           A[i * 4 + 2] * x[2] + A[i * 4 + 3] * x[3];
}
__device__ __forceinline__ void inv4(const float* m, float* o) {
  float i0  =  m[5]*m[10]*m[15] - m[5]*m[11]*m[14] - m[9]*m[6]*m[15] + m[9]*m[7]*m[14] + m[13]*m[6]*m[11] - m[13]*m[7]*m[10];
  float i4  = -m[4]*m[10]*m[15] + m[4]*m[11]*m[14] + m[8]*m[6]*m[15] - m[8]*m[7]*m[14] - m[12]*m[6]*m[11] + m[12]*m[7]*m[10];
  float i8  =  m[4]*m[ 9]*m[15] - m[4]*m[11]*m[13] - m[8]*m[5]*m[15] + m[8]*m[7]*m[13] + m[12]*m[5]*m[11] - m[12]*m[7]*m[ 9];
  float i12 = -m[4]*m[ 9]*m[14] + m[4]*m[10]*m[13] + m[8]*m[5]*m[14] - m[8]*m[6]*m[13] - m[12]*m[5]*m[10] + m[12]*m[6]*m[ 9];
  float i1  = -m[1]*m[10]*m[15] + m[1]*m[11]*m[14] + m[9]*m[2]*m[15] - m[9]*m[3]*m[14] - m[13]*m[2]*m[11] + m[13]*m[3]*m[10];
  float i5  =  m[0]*m[10]*m[15] - m[0]*m[11]*m[14] - m[8]*m[2]*m[15] + m[8]*m[3]*m[14] + m[12]*m[2]*m[11] - m[12]*m[3]*m[10];
  float i9  = -m[0]*m[ 9]*m[15] + m[0]*m[11]*m[13] + m[8]*m[1]*m[15] - m[8]*m[3]*m[13] - m[12]*m[1]*m[11] + m[12]*m[3]*m[ 9];
  float i13 =  m[0]*m[ 9]*m[14] - m[0]*m[10]*m[13] - m[8]*m[1]*m[14] + m[8]*m[2]*m[13] + m[12]*m[1]*m[10] - m[12]*m[2]*m[ 9];
  float i2  =  m[1]*m[ 6]*m[15] - m[1]*m[ 7]*m[14] - m[5]*m[2]*m[15] + m[5]*m[3]*m[14] + m[13]*m[2]*m[ 7] - m[13]*m[3]*m[ 6];
  float i6  = -m[0]*m[ 6]*m[15] + m[0]*m[ 7]*m[14] + m[4]*m[2]*m[15] - m[4]*m[3]*m[14] - m[12]*m[2]*m[ 7] + m[12]*m[3]*m[ 6];
  float i10 =  m[0]*m[ 5]*m[15] - m[0]*m[ 7]*m[13] - m[4]*m[1]*m[15] + m[4]*m[3]*m[13] + m[12]*m[1]*m[ 7] - m[12]*m[3]*m[ 5];
  float i14 = -m[0]*m[ 5]*m[14] + m[0]*m[ 6]*m[13] + m[4]*m[1]*m[14] - m[4]*m[2]*m[13] - m[12]*m[1]*m[ 6] + m[12]*m[2]*m[ 5];
  float i3  = -m[1]*m[ 6]*m[11] + m[1]*m[ 7]*m[10] + m[5]*m[2]*m[11] - m[5]*m[3]*m[10] - m[9]*m[2]*m[ 7] + m[9]*m[3]*m[ 6];
  float i7  =  m[0]*m[ 6]*m[11] - m[0]*m[ 7]*m[10] - m[4]*m[2]*m[11] + m[4]*m[3]*m[10] + m[8]*m[2]*m[ 7] - m[8]*m[3]*m[ 6];
  float i11 = -m[0]*m[ 5]*m[11] + m[0]*m[ 7]*m[ 9] + m[4]*m[1]*m[11] - m[4]*m[3]*m[ 9] - m[8]*m[1]*m[ 7] + m[8]*m[3]*m[ 5];
  float i15 =  m[0]*m[ 5]*m[10] - m[0]*m[ 6]*m[ 9] - m[4]*m[1]*m[10] + m[4]*m[2]*m[ 9] + m[8]*m[1]*m[ 6] - m[8]*m[2]*m[ 5];
  const float det = m[0]*i0 + m[1]*i4 + m[2]*i8 + m[3]*i12;
  const float rd = fast_rcp(det);          // v_rcp_f32, not IEEE div chain
  o[0]=i0*rd;  o[1]=i1*rd;  o[2]=i2*rd;  o[3]=i3*rd;
  o[4]=i4*rd;  o[5]=i5*rd;  o[6]=i6*rd;  o[7]=i7*rd;
  o[8]=i8*rd;  o[9]=i9*rd;  o[10]=i10*rd; o[11]=i11*rd;
  o[12]=i12*rd; o[13]=i13*rd; o[14]=i14*rd; o[15]=i15*rd;
}

// ---------------------------------------------------------------------------
// Kernel 2: fused forward filter (information form) + backward RTS smoother.
// One thread per batch element b (scan over T is inherently sequential).
// ---------------------------------------------------------------------------
__global__ void kvae_filter_smooth(const float* __restrict__ a,       // (RT,20)
                                   const float* __restrict__ mixed,   // (RT,96)
                                   float* __restrict__ zhat_ws,       // (RT,4)
                                   float* __restrict__ shat_ws,       // (RT,16)
                                   float* __restrict__ out_z,         // (RT,4)
                                   float* __restrict__ out_s) {       // (RT,16)
  const int b = blockIdx.x * blockDim.x + threadIdx.x;
  if (b >= KB) return;

  const float inv_r = 1.0f / 0.03f;        // exact compile-time constant

  float z[4] = {0.0f, 0.0f, 0.0f, 0.0f};
  float P[16];
#pragma unroll
  for (int i = 0; i < 16; ++i) P[i] = 0.0f;
  P[0] = P[5] = P[10] = P[15] = 20.0f;     // std0 = 20*I

  // ---------------- forward filter ----------------
  for (int t = 0; t < KT; ++t) {
    const size_t rt = (size_t)b * KT + (size_t)t;
    const float* MX = mixed + rt * 96;
    const float* arow = a + rt * 20;
    if (t + 1 < KT) {                      // hide next step's streams (global_prefetch_b8)
      __builtin_prefetch(mixed + (rt + 1) * 96, 0, 1);
      __builtin_prefetch(a + (rt + 1) * 20, 0, 1);
    }

    float Ai[16];
#pragma unroll
    for (int i = 0; i < 16; ++i) Ai[i] = MX[i];

    // z_hat = Ai z ; P_hat = Ai P Ai^T + 0.08 I
    float zh[4];  matvec4(Ai, z, zh);
    float T1[16]; mul4(Ai, P, T1);
    float Ph[16]; mul4_ABt(T1, Ai, Ph);
    Ph[0] += 0.08f; Ph[5] += 0.08f; Ph[10] += 0.08f; Ph[15] += 0.08f;

    // stored (smoother) version: elementwise max with 1e-4*I
    {
      float sc[16];
#pragma unroll
      for (int i = 0; i < 4; ++i)
#pragma unroll
        for (int j = 0; j < 4; ++j)
          sc[i * 4 + j] = fmaxf(Ph[i * 4 + j], (i == j) ? 1e-4f : 0.0f);
#pragma unroll
      for (int i = 0; i < 16; ++i) shat_ws[rt * 16 + i] = sc[i];
#pragma unroll
      for (int i = 0; i < 4; ++i) zhat_ws[rt * 4 + i] = zh[i];
    }

    // stream C rows: accumulate C^T C (sym, 10) and C^T (a - C z_hat)
    float ctc[10] = {0,0,0,0,0,0,0,0,0,0};
    float ctr[4]  = {0,0,0,0};
    const float4* crow = reinterpret_cast<const float4*>(MX + 16);
#pragma unroll
    for (int r = 0; r < KDA; ++r) {
      const float4 c = crow[r];
      const float resid = arow[r] - (c.x*zh[0] + c.y*zh[1] + c.z*zh[2] + c.w*zh[3]);
      ctr[0] += c.x * resid; ctr[1] += c.y * resid;
      ctr[2] += c.z * resid; ctr[3] += c.w * resid;
      ctc[0] += c.x*c.x; ctc[1] += c.x*c.y; ctc[2] += c.x*c.z; ctc[3] += c.x*c.w;
      ctc[4] += c.y*c.y; ctc[5] += c.y*c.z; ctc[6] += c.y*c.w;
      ctc[7] += c.z*c.z; ctc[8] += c.z*c.w; ctc[9] += c.w*c.w;
    }
    float CtC[16] = {ctc[0],ctc[1],ctc[2],ctc[3],
                     ctc[1],ctc[4],ctc[5],ctc[6],
                     ctc[2],ctc[5],ctc[7],ctc[8],
                     ctc[3],ctc[6],ctc[8],ctc[9]};

    // information form: M = (P_hat^-1 + C^T C / r)^-1  (Woodbury; R = r*I)
    float Pinv[16]; inv4(Ph, Pinv);
    float N[16];
#pragma unroll
    for (int i = 0; i < 16; ++i) N[i] = Pinv[i] + inv_r * CtC[i];
    float M[16]; inv4(N, M);

    // post_z = z_hat + M (C^T resid) / r          [Kg resid without forming Kg]
    float w[4]; matvec4(M, ctr, w);
    float pz[4];
#pragma unroll
    for (int i = 0; i < 4; ++i) pz[i] = zh[i] + inv_r * w[i];

    // IKC = I - Kg C = I - M CtC / r ;  Kg R Kg^T = (M CtC M^T)/r = G M^T / r
    float G[16]; mul4(M, CtC, G);
    float IKC[16];
#pragma unroll
    for (int i = 0; i < 16; ++i) IKC[i] = -inv_r * G[i];
    IKC[0] += 1.0f; IKC[5] += 1.0f; IKC[10] += 1.0f; IKC[15] += 1.0f;

    float T2[16]; mul4(IKC, Ph, T2);
    float PS[16]; mul4_ABt(T2, IKC, PS);      // IKC P_hat IKC^T
    float T3[16]; mul4_ABt(G, M, T3);         // G M^T
#pragma unroll
    for (int i = 0; i < 16; ++i) PS[i] += inv_r * T3[i];

    // posteriors go straight into d_out (smoother updates in place)
#pragma unroll
    for (int i = 0; i < 4; ++i)  out_z[rt * 4 + i] = pz[i];
#pragma unroll
    for (int i = 0; i < 16; ++i) out_s[rt * 16 + i] = PS[i];

#pragma unroll
    for (int i = 0; i < 4; ++i)  z[i] = pz[i];
#pragma unroll
    for (int i = 0; i < 16; ++i) P[i] = PS[i];
  }

  // ---------------- backward RTS smoother ----------------
  // carry = posterior at T-1 (already the smoothed value, already in d_out)
  float zc[4];
#pragma unroll
  for (int i = 0; i < 4; ++i) zc[i] = z[i];
  float Sc[16];
#pragma unroll
  for (int i = 0; i < 16; ++i) Sc[i] = P[i];

  for (int t = KT - 2; t >= 0; --t) {
    const size_t rt  = (size_t)b * KT + (size_t)t;
    const size_t rt1 = rt + 1;

    float Pt[16], pzt[4], A1[16], Sh1[16], zh1[4];
#pragma unroll
    for (int i = 0; i < 16; ++i) Pt[i]  = out_s[rt * 16 + i];
#pragma unroll
    for (int i = 0; i < 4; ++i)  pzt[i] = out_z[rt * 4 + i];
#pragma unroll
    for (int i = 0; i < 16; ++i) A1[i]  = mixed[rt1 * 96 + i];
#pragma unroll
    for (int i = 0; i < 16; ++i) Sh1[i] = shat_ws[rt1 * 16 + i];
#pragma unroll
    for (int i = 0; i < 4; ++i)  zh1[i] = zhat_ws[rt1 * 4 + i];

    // D = post_std_t A_{t+1}^T inv(std_hat_c_{t+1})
    float ShInv[16]; inv4(Sh1, ShInv);
    float T2[16]; mul4_ABt(Pt, A1, T2);
    float D[16];  mul4(T2, ShInv, D);

    float dz[4];
#pragma unroll
    for (int i = 0; i < 4; ++i) dz[i] = zc[i] - zh1[i];
    float Ddz[4]; matvec4(D, dz, Ddz);
    float zs[4];
#pragma unroll
    for (int i = 0; i < 4; ++i) zs[i] = pzt[i] + Ddz[i];

    float dS[16];
#pragma unroll
    for (int i = 0; i < 16; ++i) dS[i] = Sc[i] - Sh1[i];
    float T3[16]; mul4(D, dS, T3);
    float Ss[16]; mul4_ABt(T3, D, Ss);
#pragma unroll
    for (int i = 0; i < 16; ++i) Ss[i] += Pt[i];

#pragma unroll
    for (int i = 0; i < 4; ++i)  out_z[rt * 4 + i] = zs[i];
#pragma unroll
    for (int i = 0; i < 16; ++i) out_s[rt * 16 + i] = Ss[i];

#pragma unroll
    for (int i = 0; i < 4; ++i)  zc[i] = zs[i];
#pragma unroll
    for (int i = 0; i < 16; ++i) Sc[i] = Ss[i];
  }
}

// ---------------------------------------------------------------------------
extern "C" void kernel_launch(void* const* d_in, const int* in_sizes, int n_in,
                              void* d_out, int out_size, void* d_ws, size_t ws_size,
                              hipStream_t stream) {
  const float* a            = (const float*)d_in[0]; // (B,T,20)
  const float* alpha_logits = (const float*)d_in[1]; // (B,T,3)
  const float* A_base       = (const float*)d_in[2]; // (3,4,4)
  const float* C_base       = (const float*)d_in[3]; // (3,20,4)

  // workspace carve-up (needs RT*116 floats ~ 244 MB)
  float* mixed   = (float*)d_ws;                 // RT*96
  float* zhat_ws = mixed + KRT * 96;             // RT*4
  float* shat_ws = zhat_ws + KRT * 4;            // RT*16

  float* out_z = (float*)d_out;                  // RT*4
  float* out_s = out_z + KRT * 4;                // RT*16

  // Kernel 1: 16 rows per wave, 8 waves per block -> RT/128 blocks
  {
    dim3 block(256);
    dim3 grid((unsigned)(KRT / 128));            // 524288/128 = 4096
    kvae_mix_wmma<<<grid, block, 0, stream>>>(alpha_logits, A_base, C_base, mixed);
  }

  // Kernel 2: one thread per batch element; small blocks to spread across WGPs
  {
    dim3 block(64);
    dim3 grid(KB / 64);                          // 32 blocks
    kvae_filter_smooth<<<grid, block, 0, stream>>>(a, mixed, zhat_ws, shat_ws,
                                                   out_z, out_s);
  }
}